// ArmModel_76811195122292
// MI455X (gfx1250) — compile-verified
//
#include <hip/hip_runtime.h>
#include <cstdint>

#define NDOF 7
#define DT_F 0.1f
#define MAX_VEL_F 20.0f
#define ACT_RANGE_F 50.0f

// ---- static LDS layout (float indices) : shared constants ----
#define L_MP   0     // 128 : M_param raw (8 x 4 x 4)
#define L_A    128   // 42  : A (7 x 6)
#define L_G    170   // 28  : G_param (7 x 4)
#define L_GRAV 198   // 3   : gravity
#define L_FTIP 201   // 6   : ftip
#define L_MINV 208   // 8*12 : per-link trans_inv {R^T (9), -R^T p (3)}
#define L_TOT  304

// per-thread dynamic-LDS slot: 7 joints x {Vs(6), Vd(6)} interleaved = 84 floats.
// stride 86: 344 B (8B aligned -> b64 merge ok); 22*L mod 64 distinct for 32 lanes.
#define TSLOT  86

#define TRI(i, j) (((i) * ((i) + 1)) / 2 + (j))   // lower-triangle index, i >= j

// ---------------- small dense helpers (register-resident after SROA) ----
__device__ __forceinline__ void mat33mul(const float* A, const float* B, float* C) {
#pragma unroll
  for (int r = 0; r < 3; ++r)
#pragma unroll
    for (int c = 0; c < 3; ++c)
      C[r*3+c] = A[r*3+0]*B[0+c] + A[r*3+1]*B[3+c] + A[r*3+2]*B[6+c];
}
__device__ __forceinline__ void matvec3(const float* R, const float* x, float* y) {
  y[0] = R[0]*x[0] + R[1]*x[1] + R[2]*x[2];
  y[1] = R[3]*x[0] + R[4]*x[1] + R[5]*x[2];
  y[2] = R[6]*x[0] + R[7]*x[1] + R[8]*x[2];
}
__device__ __forceinline__ void matTvec3(const float* R, const float* x, float* y) {
  y[0] = R[0]*x[0] + R[3]*x[1] + R[6]*x[2];
  y[1] = R[1]*x[0] + R[4]*x[1] + R[7]*x[2];
  y[2] = R[2]*x[0] + R[5]*x[1] + R[8]*x[2];
}
__device__ __forceinline__ void cross3(const float* a, const float* b, float* c) {
  c[0] = a[1]*b[2] - a[2]*b[1];
  c[1] = a[2]*b[0] - a[0]*b[2];
  c[2] = a[0]*b[1] - a[1]*b[0];
}
// out = Ad(R,p) * (w,v) :  w' = R w ; v' = p x w' + R v
__device__ __forceinline__ void adj_apply(const float* R, const float* p,
                                          const float* w, const float* v,
                                          float* ow, float* ov) {
  matvec3(R, w, ow);
  float pc[3]; cross3(p, ow, pc);
  float rv[3]; matvec3(R, v, rv);
  ov[0] = pc[0] + rv[0]; ov[1] = pc[1] + rv[1]; ov[2] = pc[2] + rv[2];
}
// out = Ad(R,p)^T * (Fw,Fv) : Fw' = R^T (Fw - p x Fv) ; Fv' = R^T Fv
__device__ __forceinline__ void adjT_apply(const float* R, const float* p,
                                           const float* Fw, const float* Fv,
                                           float* ow, float* ov) {
  float pc[3]; cross3(p, Fv, pc);
  float t[3] = { Fw[0]-pc[0], Fw[1]-pc[1], Fw[2]-pc[2] };
  matTvec3(R, t, ow);
  matTvec3(R, Fv, ov);
}
// exp6 of twist xi=(w,v) -> affine (R,p); matches reference incl. small-angle path
__device__ __forceinline__ void exp6_affine(float wx, float wy, float wz,
                                            float vx, float vy, float vz,
                                            float* R, float* p) {
  float th = sqrtf(wx*wx + wy*wy + wz*wz);
  bool small = th < 1e-10f;
  float inv = small ? 1.0f : (1.0f / th);
  float ux = wx*inv, uy = wy*inv, uz = wz*inv;
  float s = __sinf(th), c = __cosf(th);
  float a = 1.0f - c;
  R[0] = c + a*ux*ux;    R[1] = a*ux*uy - s*uz; R[2] = a*ux*uz + s*uy;
  R[3] = a*uy*ux + s*uz; R[4] = c + a*uy*uy;    R[5] = a*uy*uz - s*ux;
  R[6] = a*uz*ux - s*uy; R[7] = a*uz*uy + s*ux; R[8] = c + a*uz*uz;
  float tx = vx*inv, ty = vy*inv, tz = vz*inv;        // v / th_s
  float d = ux*tx + uy*ty + uz*tz;
  float cx = uy*tz - uz*ty, cy = uz*tx - ux*tz, cz = ux*ty - uy*tx;
  float bcoef = th - s;
  float px = th*tx + a*cx + bcoef*(ux*d - tx);
  float py = th*ty + a*cy + bcoef*(uy*d - ty);
  float pz = th*tz + a*cz + bcoef*(uz*d - tz);
  if (small) {
    R[0]=1.f;R[1]=0.f;R[2]=0.f;R[3]=0.f;R[4]=1.f;R[5]=0.f;R[6]=0.f;R[7]=0.f;R[8]=1.f;
    px = vx; py = vy; pz = vz;
  }
  p[0] = px; p[1] = py; p[2] = pz;
}

extern __shared__ float sDyn[];   // 256 * TSLOT floats, per-thread Vs/Vd history

__global__ void __launch_bounds__(256)
arm_rk4_kernel(const float* __restrict__ state, const float* __restrict__ torque,
               const float* __restrict__ Mp,    const float* __restrict__ Aarr,
               const float* __restrict__ Gp,    const float* __restrict__ grav,
               const float* __restrict__ ftip,  float* __restrict__ out, int batch)
{
  __shared__ float sC[L_TOT];
  const int tid = threadIdx.x;

  // ---- Stage the 207 shared parameter floats into LDS via CDNA5 async DMA ----
  if (tid < 207) {   // masked lanes of wave 6 simply don't issue (per-lane async)
    const float* src; int li;
    if      (tid < 128) { src = Mp   + tid;        li = L_MP + tid; }
    else if (tid < 170) { src = Aarr + (tid-128);  li = tid; }
    else if (tid < 198) { src = Gp   + (tid-170);  li = tid; }
    else if (tid < 201) { src = grav + (tid-198);  li = tid; }
    else                { src = ftip + (tid-201);  li = tid; }
    uint32_t lds_addr = (uint32_t)(uintptr_t)(&sC[li]);
    uint64_t gaddr    = (uint64_t)(uintptr_t)src;
    asm volatile("global_load_async_to_lds_b32 %0, %1, off"
                 :: "v"(lds_addr), "v"(gaddr) : "memory");
    asm volatile("s_wait_asynccnt 0x0" ::: "memory");
  }
  __syncthreads();

  // ---- trans_inv(Mlist[i]) for i=0..7 once per block (R^T, -R^T p) ----
  if (tid < 8) {
    const float* Mi = &sC[L_MP + tid*16];
    float r00=Mi[0], r01=Mi[1], r02=Mi[2],  p0=Mi[3];
    float r10=Mi[4], r11=Mi[5], r12=Mi[6],  p1=Mi[7];
    float r20=Mi[8], r21=Mi[9], r22=Mi[10], p2=Mi[11];
    float* o = &sC[L_MINV + tid*12];
    o[0]=r00; o[1]=r10; o[2]=r20;
    o[3]=r01; o[4]=r11; o[5]=r21;
    o[6]=r02; o[7]=r12; o[8]=r22;
    o[9]  = -(r00*p0 + r10*p1 + r20*p2);
    o[10] = -(r01*p0 + r11*p1 + r21*p2);
    o[11] = -(r02*p0 + r12*p1 + r22*p2);
  }
  __syncthreads();

  const int b = blockIdx.x * blockDim.x + tid;
  if (b >= batch) return;

  float* tvs = sDyn + tid * TSLOT;   // joint i: [i*12..+5] = Vs_i, [i*12+6..+11] = Vd_i

  // ---- vectorized per-element input: state row is 8B aligned (b*56 bytes) ----
  float q0[7], d0[7], qf[7];
  {
    const float2* srow =
        (const float2*)__builtin_assume_aligned(state + (size_t)b*14, 8);
    float sv[14];
#pragma unroll
    for (int i = 0; i < 7; ++i) {
      float2 e = srow[i];
      sv[2*i] = e.x; sv[2*i+1] = e.y;
    }
#pragma unroll
    for (int i = 0; i < 7; ++i) { q0[i] = sv[i]; d0[i] = sv[7+i]; }
#pragma unroll
    for (int i = 0; i < 7; ++i) qf[i] = torque[b*7 + i] * ACT_RANGE_F;
  }
  const float gx = sC[L_GRAV+0], gy = sC[L_GRAV+1], gz = sC[L_GRAV+2];

  float pq[7], pd[7], accq[7], accd[7];
#pragma unroll
  for (int i = 0; i < 7; ++i) { pq[i]=q0[i]; pd[i]=d0[i]; accq[i]=0.f; accd[i]=0.f; }

  // ---------------- RK4: 4 stages, derivs body kept single-copy -------------
#pragma unroll 1
  for (int st = 0; st < 4; ++st) {
    // (1) joint adjoints Ti = exp6(-A_i q_i) * trans_inv(M_i); shared by all work below
    float Rj[7][9], pj[7][3];
#pragma unroll
    for (int i = 0; i < 7; ++i) {
      const float* Ai = &sC[L_A + i*6];
      float qi = pq[i];
      float Re[9], pe[3];
      exp6_affine(-Ai[0]*qi, -Ai[1]*qi, -Ai[2]*qi,
                  -Ai[3]*qi, -Ai[4]*qi, -Ai[5]*qi, Re, pe);
      const float* Mi = &sC[L_MINV + i*12];
      mat33mul(Re, Mi, Rj[i]);
      pj[i][0] = Re[0]*Mi[9] + Re[1]*Mi[10] + Re[2]*Mi[11] + pe[0];
      pj[i][1] = Re[3]*Mi[9] + Re[4]*Mi[10] + Re[5]*Mi[11] + pe[1];
      pj[i][2] = Re[6]*Mi[9] + Re[7]*Mi[10] + Re[8]*Mi[11] + pe[2];
    }

    // (2) bias force h = ID(q, dq, ddq=0, g, ftip); V/Vd history in LDS
    float rhs[7];
    {
      float vw[3] = {0,0,0}, vv[3] = {0,0,0};
      float dw[3] = {0,0,0}, dv[3] = {-gx, -gy, -gz};
#pragma unroll
      for (int i = 0; i < 7; ++i) {
        const float* Ai = &sC[L_A + i*6];
        float dqi = pd[i];
        float nw[3], nv[3];
        adj_apply(Rj[i], pj[i], vw, vv, nw, nv);
        nw[0]+=Ai[0]*dqi; nw[1]+=Ai[1]*dqi; nw[2]+=Ai[2]*dqi;
        nv[0]+=Ai[3]*dqi; nv[1]+=Ai[4]*dqi; nv[2]+=Ai[5]*dqi;
        float mw[3], mv[3];
        adj_apply(Rj[i], pj[i], dw, dv, mw, mv);
        float c1[3], c2a[3], c2b[3];
        cross3(nw, Ai,   c1);   // w x Aw
        cross3(nv, Ai,   c2a);  // v x Aw
        cross3(nw, Ai+3, c2b);  // w x Av
#pragma unroll
        for (int k = 0; k < 3; ++k) {
          mw[k] += c1[k]*dqi;
          mv[k] += (c2a[k] + c2b[k])*dqi;
          vw[k] = nw[k]; vv[k] = nv[k]; dw[k] = mw[k]; dv[k] = mv[k];
          tvs[i*12 + k]     = nw[k];  tvs[i*12 + 3 + k] = nv[k];   // Vs_i
          tvs[i*12 + 6 + k] = mw[k];  tvs[i*12 + 9 + k] = mv[k];   // Vd_i
        }
      }
      float fw[3] = { sC[L_FTIP+0], sC[L_FTIP+1], sC[L_FTIP+2] };
      float fv[3] = { sC[L_FTIP+3], sC[L_FTIP+4], sC[L_FTIP+5] };
#pragma unroll
      for (int i = 6; i >= 0; --i) {
        float tw[3], tv[3];
        if (i == 6) adjT_apply(&sC[L_MINV + 7*12], &sC[L_MINV + 7*12 + 9], fw, fv, tw, tv);
        else        adjT_apply(Rj[i+1], pj[i+1], fw, fv, tw, tv);
        const float* Gi = &sC[L_G + i*4];
        float vwi[3] = { tvs[i*12+0], tvs[i*12+1], tvs[i*12+2] };
        float vvi[3] = { tvs[i*12+3], tvs[i*12+4], tvs[i*12+5] };
        float dwi[3] = { tvs[i*12+6], tvs[i*12+7], tvs[i*12+8] };
        float dvi[3] = { tvs[i*12+9], tvs[i*12+10], tvs[i*12+11] };
        float gvw[3] = { Gi[0]*vwi[0], Gi[1]*vwi[1], Gi[2]*vwi[2] };
        float gvv[3] = { Gi[3]*vvi[0], Gi[3]*vvi[1], Gi[3]*vvi[2] };
        float ca[3], cb[3], cc[3];
        cross3(vwi, gvw, ca); cross3(vvi, gvv, cb); cross3(vwi, gvv, cc);
#pragma unroll
        for (int k = 0; k < 3; ++k) {
          fw[k] = tw[k] + Gi[k]*dwi[k] + ca[k] + cb[k];
          fv[k] = tv[k] + Gi[3]*dvi[k] + cc[k];
        }
        const float* Ai = &sC[L_A + i*6];
        float hi = fw[0]*Ai[0]+fw[1]*Ai[1]+fw[2]*Ai[2]
                 + fv[0]*Ai[3]+fv[1]*Ai[4]+fv[2]*Ai[5];
        rhs[i] = qf[i] - hi;            // fold h directly into the RHS
      }
    }

    // (3) mass matrix: single ascending sweep, M = J^T diag(G) J.
    //     Entry (j,k) first touched at level i==j -> assign, then accumulate.
    float Mc[28];
    {
      float Jw[7][3], Jv[7][3];
#pragma unroll
      for (int i = 0; i < 7; ++i) {
        // propagate existing columns j<i through AdT_i
#pragma unroll
        for (int j = 0; j < i; ++j) {
          float tw[3], tv[3];
          adj_apply(Rj[i], pj[i], Jw[j], Jv[j], tw, tv);
          Jw[j][0]=tw[0]; Jw[j][1]=tw[1]; Jw[j][2]=tw[2];
          Jv[j][0]=tv[0]; Jv[j][1]=tv[1]; Jv[j][2]=tv[2];
        }
        // new column i enters as A_i
        const float* Ai = &sC[L_A + i*6];
        Jw[i][0]=Ai[0]; Jw[i][1]=Ai[1]; Jw[i][2]=Ai[2];
        Jv[i][0]=Ai[3]; Jv[i][1]=Ai[4]; Jv[i][2]=Ai[5];
        // M_jk (+)= (G_i o J_j) . J_k for all j,k <= i
        const float* Gi = &sC[L_G + i*4];
#pragma unroll
        for (int j = 0; j <= i; ++j) {
          float gw0 = Gi[0]*Jw[j][0], gw1 = Gi[1]*Jw[j][1], gw2 = Gi[2]*Jw[j][2];
          float gv0 = Gi[3]*Jv[j][0], gv1 = Gi[3]*Jv[j][1], gv2 = Gi[3]*Jv[j][2];
#pragma unroll
          for (int k = 0; k <= j; ++k) {
            float dot = gw0*Jw[k][0] + gw1*Jw[k][1] + gw2*Jw[k][2]
                      + gv0*Jv[k][0] + gv1*Jv[k][1] + gv2*Jv[k][2];
            if (i == j) Mc[TRI(j,k)] = dot;       // first touch
            else        Mc[TRI(j,k)] += dot;
          }
        }
      }
    }

    // (4) LDL^T factor + solve (SPD mass matrix), compact triangle
    float Dinv[7], qacc[7];
    {
      float Dg[7];
#pragma unroll
      for (int j = 0; j < 7; ++j) {
        float dj = Mc[TRI(j,j)];
#pragma unroll
        for (int k = 0; k < j; ++k) dj -= Mc[TRI(j,k)] * Mc[TRI(j,k)] * Dg[k];
        Dg[j] = dj;
        float inv = 1.0f / dj;
        Dinv[j] = inv;
#pragma unroll
        for (int i = j+1; i < 7; ++i) {
          float s = Mc[TRI(i,j)];
#pragma unroll
          for (int k = 0; k < j; ++k) s -= Mc[TRI(i,k)] * Mc[TRI(j,k)] * Dg[k];
          Mc[TRI(i,j)] = s * inv;
        }
      }
    }
#pragma unroll
    for (int i = 0; i < 7; ++i) {           // L y = rhs
      float s = rhs[i];
#pragma unroll
      for (int k = 0; k < i; ++k) s -= Mc[TRI(i,k)] * rhs[k];
      rhs[i] = s;
    }
#pragma unroll
    for (int i = 0; i < 7; ++i) rhs[i] *= Dinv[i];   // D z = y
#pragma unroll
    for (int i = 6; i >= 0; --i) {          // L^T x = z
      float s = rhs[i];
#pragma unroll
      for (int k = i+1; k < 7; ++k) s -= Mc[TRI(k,i)] * qacc[k];
      qacc[i] = s;
    }

    // RK4 accumulate + next evaluation point
    float wgt   = (st == 0 || st == 3) ? 1.0f : 2.0f;
    float alpha = (st < 2) ? (0.5f * DT_F) : DT_F;
#pragma unroll
    for (int i = 0; i < 7; ++i) {
      accq[i] += wgt * pd[i];
      accd[i] += wgt * qacc[i];
      float nq = q0[i] + alpha * pd[i];
      float nd = d0[i] + alpha * qacc[i];
      pq[i] = nq; pd[i] = nd;
    }
  }

  // ---------------- finalize state: wrap q, clip dq; vectorized store -------
  const float PI_F  = 3.14159265358979323846f;
  const float TPI_F = 6.28318530717958647692f;
  float q1[7], ov[14];
#pragma unroll
  for (int i = 0; i < 7; ++i) {
    float q = q0[i] + (DT_F / 6.0f) * accq[i];
    float d = d0[i] + (DT_F / 6.0f) * accd[i];
    float r = fmodf(q + PI_F, TPI_F);
    r = (r < 0.0f) ? r + TPI_F : r;
    q1[i] = r - PI_F;
    ov[i]     = q1[i];
    ov[7 + i] = fminf(fmaxf(d, -MAX_VEL_F), MAX_VEL_F);
  }
  {
    float2* orow = (float2*)__builtin_assume_aligned(out + (size_t)b*14, 8);
#pragma unroll
    for (int i = 0; i < 7; ++i) { float2 e; e.x = ov[2*i]; e.y = ov[2*i+1]; orow[i] = e; }
  }

  // ---------------- forward kinematics: ee = (Π M_i E_i · M_7).p ----------
  float Rt_[9] = {1,0,0, 0,1,0, 0,0,1};
  float pt_[3] = {0,0,0};
#pragma unroll
  for (int i = 0; i < 7; ++i) {
    const float* Mi = &sC[L_MP + i*16];
    float Rm[9] = { Mi[0],Mi[1],Mi[2], Mi[4],Mi[5],Mi[6], Mi[8],Mi[9],Mi[10] };
    float pm[3] = { Mi[3], Mi[7], Mi[11] };
    float RT[9], pT[3], t0[3];
    mat33mul(Rt_, Rm, RT);
    matvec3(Rt_, pm, t0);
    pT[0]=t0[0]+pt_[0]; pT[1]=t0[1]+pt_[1]; pT[2]=t0[2]+pt_[2];
    const float* Ai = &sC[L_A + i*6];
    float qi = q1[i];
    float Re[9], pe[3];
    exp6_affine(Ai[0]*qi, Ai[1]*qi, Ai[2]*qi, Ai[3]*qi, Ai[4]*qi, Ai[5]*qi, Re, pe);
    mat33mul(RT, Re, Rt_);
    float t1[3]; matvec3(RT, pe, t1);
    pt_[0]=t1[0]+pT[0]; pt_[1]=t1[1]+pT[1]; pt_[2]=t1[2]+pT[2];
  }
  {
    const float* M7 = &sC[L_MP + 7*16];
    float pm[3] = { M7[3], M7[7], M7[11] };
    float t[3]; matvec3(Rt_, pm, t);
    out[batch*14 + b*3 + 0] = t[0] + pt_[0];
    out[batch*14 + b*3 + 1] = t[1] + pt_[1];
    out[batch*14 + b*3 + 2] = t[2] + pt_[2];
  }
}

extern "C" void kernel_launch(void* const* d_in, const int* in_sizes, int n_in,
                              void* d_out, int out_size, void* d_ws, size_t ws_size,
                              hipStream_t stream) {
  (void)n_in; (void)out_size; (void)d_ws; (void)ws_size;
  const float* state  = (const float*)d_in[0];
  const float* torque = (const float*)d_in[1];
  const float* Mp     = (const float*)d_in[2];
  const float* Aarr   = (const float*)d_in[3];
  const float* Gp     = (const float*)d_in[4];
  const float* grav   = (const float*)d_in[5];
  const float* ftipv  = (const float*)d_in[6];
  float* out = (float*)d_out;
  int batch = in_sizes[0] / 14;
  dim3 block(256), grid((batch + 255) / 256);
  size_t shmem = (size_t)256 * TSLOT * sizeof(float);   // per-thread V/Vd history
  arm_rk4_kernel<<<grid, block, shmem, stream>>>(state, torque, Mp, Aarr, Gp, grav,
                                                 ftipv, out, batch);
}